// DFR_23673859736399
// MI455X (gfx1250) — compile-verified
//
#include <hip/hip_runtime.h>

typedef float v2f __attribute__((ext_vector_type(2)));
typedef float v8f __attribute__((ext_vector_type(8)));

constexpr int Bn = 16, Hh = 64, Ww = 64, Cc = 512;
constexpr int HW = Hh * Ww;          // 4096
constexpr int RB = 64;               // partial blocks per batch
constexpr int ROWS = HW / RB;        // 64 rows per block

__device__ __forceinline__ float silu_f(float v) { return v / (1.0f + __expf(-v)); }
__device__ __forceinline__ float softplus_f(float v) {
    return (v > 20.0f) ? v : log1pf(__expf(v));
}

// ---------------- Kernel 1: partial column sums of x1 ----------------
__global__ void __launch_bounds__(256) gap_partial(const float* __restrict__ x1,
                                                   float* __restrict__ partial) {
    const int blk = blockIdx.x;
    const int b = blk / RB, rb = blk % RB;
    const int tid = threadIdx.x;
    const float* base = x1 + ((size_t)b * HW + (size_t)rb * ROWS) * Cc;
    float s0 = 0.f, s1 = 0.f;
    for (int r = 0; r < ROWS; ++r) {
        s0 += base[(size_t)r * Cc + tid];
        s1 += base[(size_t)r * Cc + tid + 256];
    }
    float* p = partial + ((size_t)b * RB + rb) * Cc;
    p[tid] = s0;
    p[tid + 256] = s1;
}

// ---------------- Kernel 2: reduce partials -> pooled mean ----------------
__global__ void __launch_bounds__(256) gap_reduce(const float* __restrict__ partial,
                                                  float* __restrict__ pooled) {
    const int gid = blockIdx.x * 256 + threadIdx.x;   // 0..Bn*Cc-1
    const int b = gid / Cc, c = gid % Cc;
    const float* p = partial + (size_t)b * RB * Cc + c;
    float s = 0.f;
    for (int rb = 0; rb < RB; ++rb) s += p[(size_t)rb * Cc];
    pooled[gid] = s * (1.0f / (float)HW);
}

// ---------------- Kernel 3: tiny Mamba (per-batch block) ----------------
// LDS layout (floats):
//   s_dbl[512][36] : col 0 = dt_raw, 1..16 = B(t,s), 17..32 = C(t,s), 33..34 = dt(t,d)
//   s_x  [512][2]  : post-conv SiLU x
//   s_z  [512][2]  : gate input z
__global__ void __launch_bounds__(256) mamba_scan(
    const float* __restrict__ pooled, float* __restrict__ att,
    const float* __restrict__ ipw,   // [4][1]  in_proj_w
    const float* __restrict__ cw,    // [2][1][4] conv_w
    const float* __restrict__ cb,    // [2] conv_b
    const float* __restrict__ xpw,   // [33][2] x_proj_w
    const float* __restrict__ dtw,   // [2][1] dt_proj_w
    const float* __restrict__ dtb,   // [2] dt_proj_b
    const float* __restrict__ A_log, // [2][16]
    const float* __restrict__ Dpp,   // [2]
    const float* __restrict__ opw)   // [1][2] out_proj_w
{
    extern __shared__ float smem[];
    float* s_dbl = smem;              // 512*36
    float* s_x   = smem + 512 * 36;   // 512*2
    float* s_z   = s_x + 512 * 2;     // 512*2

    const int tid = threadIdx.x;
    const int b = blockIdx.x;
    const float* pb = pooled + (size_t)b * Cc;

    // ---- Phase A: in_proj + causal depthwise conv(k=4) + SiLU, and z ----
    for (int t = tid; t < Cc; t += 256) {
        float pv[4];
#pragma unroll
        for (int k = 0; k < 4; ++k) {
            int tk = t - 3 + k;
            pv[k] = (tk >= 0) ? pb[tk] : 0.f;
        }
#pragma unroll
        for (int d = 0; d < 2; ++d) {
            float w = ipw[d];
            float acc = cb[d];
#pragma unroll
            for (int k = 0; k < 4; ++k) acc += cw[d * 4 + k] * (pv[k] * w);
            s_x[t * 2 + d] = silu_f(acc);
            s_z[t * 2 + d] = pv[3] * ipw[2 + d];
        }
    }
    __syncthreads();

    // ---- Phase B: x_dbl = x @ x_proj_w^T via V_WMMA_F32_16X16X4_F32 ----
    // M = 16 timesteps, N = 16 output features (33 total -> 3 N-tiles), K = 4 (2 used).
    {
        const int lane = tid & 31;
        const int wave = tid >> 5;          // 8 waves
        const bool hi = lane >= 16;
        const int nl = lane & 15;
        for (int i = 0; i < 4; ++i) {
            const int mt = wave * 4 + i;    // 0..31 M-tiles
            v2f a;
            if (!hi) {                      // lanes 0-15 hold K=0,1; lanes 16-31 = K=2,3 (zero pad)
                a.x = s_x[(mt * 16 + lane) * 2 + 0];
                a.y = s_x[(mt * 16 + lane) * 2 + 1];
            } else {
                a.x = 0.f; a.y = 0.f;
            }
            for (int nt = 0; nt < 3; ++nt) {
                v2f bm;
                const int n = nt * 16 + nl;
                if (!hi && n < 33) {        // B[k][n] = x_proj_w[n][k]
                    bm.x = xpw[n * 2 + 0];
                    bm.y = xpw[n * 2 + 1];
                } else {
                    bm.x = 0.f; bm.y = 0.f;
                }
                v8f c = {};
                c = __builtin_amdgcn_wmma_f32_16x16x4_f32(
                        false, a, false, bm, (short)0, c, false, false);
                // D layout: VGPR r -> lanes 0-15: (M=r, N=lane); lanes 16-31: (M=8+r, N=lane-16)
#pragma unroll
                for (int r = 0; r < 8; ++r) {
                    const int M = hi ? (8 + r) : r;
                    const int nn = nt * 16 + nl;
                    if (nn < 33) s_dbl[(mt * 16 + M) * 36 + nn] = c[r];
                }
            }
        }
    }
    __syncthreads();

    // ---- Phase C1: dt = softplus(dt_raw * dt_proj_w + dt_proj_b) ----
    for (int t = tid; t < Cc; t += 256) {
        const float raw = s_dbl[t * 36 + 0];
        s_dbl[t * 36 + 33] = softplus_f(raw * dtw[0] + dtb[0]);
        s_dbl[t * 36 + 34] = softplus_f(raw * dtw[1] + dtb[1]);
    }
    __syncthreads();

    // ---- Phase C2: sequential selective-scan on wave 0; lane = (d,s), 2*16 = 32 lanes ----
    if (tid < 32) {
        const int lane = tid;
        const int d = lane >> 4;
        const int s = lane & 15;
        const float Ads = -__expf(A_log[d * 16 + s]);
        const float Dpd = Dpp[d];
        const float owd = opw[d];
        float h = 0.f;
        for (int t = 0; t < Cc; ++t) {
            const float* row = s_dbl + t * 36;
            const float dt = row[33 + d];
            const float Bm = row[1 + s];
            const float Cm = row[17 + s];
            const float xv = s_x[t * 2 + d];
            const float dA = __expf(dt * Ads);
            h = dA * h + dt * Bm * xv;
            float p = h * Cm;
            // reduce over s within each 16-lane half
            p += __shfl_xor(p, 1, 32);
            p += __shfl_xor(p, 2, 32);
            p += __shfl_xor(p, 4, 32);
            p += __shfl_xor(p, 8, 32);
            const float zv = s_z[t * 2 + d];
            float v = (p + xv * Dpd) * silu_f(zv) * owd;
            // combine the two d halves
            v += __shfl_xor(v, 16, 32);
            if (lane == 0) att[(size_t)b * Cc + t] = v;
        }
    }
}

// ---------------- Kernel 4: out = x2 * att (broadcast over H,W) ----------------
__global__ void __launch_bounds__(256) apply_att(const float* __restrict__ x2,
                                                 const float* __restrict__ att,
                                                 float* __restrict__ out, size_t n4) {
    const size_t i4 = (size_t)blockIdx.x * 256 + threadIdx.x;
    if (i4 >= n4) return;
    const size_t idx = i4 * 4;
    __builtin_prefetch((const char*)x2 + idx * 4 + 16384, 0, 0); // global_prefetch_b8
    const float4 xv = ((const float4*)x2)[i4];
    const size_t b = idx / ((size_t)HW * Cc);
    const int c = (int)(idx & (Cc - 1));
    const float4 av = *(const float4*)(att + b * Cc + c);
    float4 o;
    o.x = xv.x * av.x; o.y = xv.y * av.y; o.z = xv.z * av.z; o.w = xv.w * av.w;
    ((float4*)out)[i4] = o;
}

extern "C" void kernel_launch(void* const* d_in, const int* in_sizes, int n_in,
                              void* d_out, int out_size, void* d_ws, size_t ws_size,
                              hipStream_t stream) {
    const float* x1  = (const float*)d_in[0];
    const float* x2  = (const float*)d_in[1];
    const float* ipw = (const float*)d_in[2];
    const float* cw  = (const float*)d_in[3];
    const float* cb  = (const float*)d_in[4];
    const float* xpw = (const float*)d_in[5];
    const float* dtw = (const float*)d_in[6];
    const float* dtb = (const float*)d_in[7];
    const float* Al  = (const float*)d_in[8];
    const float* Dpp = (const float*)d_in[9];
    const float* opw = (const float*)d_in[10];
    float* out = (float*)d_out;

    float* ws      = (float*)d_ws;
    float* partial = ws;                                   // Bn*RB*Cc floats (2 MiB)
    float* pooled  = ws + (size_t)Bn * RB * Cc;            // Bn*Cc
    float* attb    = pooled + (size_t)Bn * Cc;             // Bn*Cc

    gap_partial<<<Bn * RB, 256, 0, stream>>>(x1, partial);
    gap_reduce<<<(Bn * Cc) / 256, 256, 0, stream>>>(partial, pooled);

    const size_t lds_bytes = (size_t)(512 * 36 + 512 * 2 + 512 * 2) * sizeof(float); // 80 KB
    mamba_scan<<<Bn, 256, lds_bytes, stream>>>(pooled, attb, ipw, cw, cb, xpw,
                                               dtw, dtb, Al, Dpp, opw);

    const size_t n4 = (size_t)Bn * HW * Cc / 4;            // 8,388,608
    apply_att<<<(unsigned)(n4 / 256), 256, 0, stream>>>(x2, attb, out, n4);
}